// RUNG_learnable_distance_11587821765105
// MI455X (gfx1250) — compile-verified
//
#include <hip/hip_runtime.h>
#include <hip/hip_bf16.h>

// ---------------------------------------------------------------------------
// Problem constants (fixed shapes per reference)
// ---------------------------------------------------------------------------
#define NN      4096
#define NW      (NN / 32)                    // 32-bit mask words per row
#define IN_DIM  512
#define HID     256
#define CC      16
#define KSTEPS  10
#define QQ      0.75f
#define SCAD_A  3.7f
#define EPS_F   1e-8f
#define LAM_F   ((float)(1.0 / 0.9 - 1.0))   // skip-connection weight lambda

typedef float v2f __attribute__((ext_vector_type(2)));
typedef float v8f __attribute__((ext_vector_type(8)));

// ---------------------------------------------------------------------------
// f32 WMMA GEMM: out[M,N] = act(A[M,K] @ B[K,N] + bias[N])   (row-major)
// One wave32 computes one 16x16 tile using V_WMMA_F32_16X16X4_F32.
// VGPR layouts per CDNA5 ISA §7.12.2:
//   A 16x4:  lanes 0-15 M=0..15 {K=k,k+1}; lanes 16-31 M=0..15 {K=k+2,k+3}
//   B 4x16:  VGPR0 rows K=k (lanes0-15)/K=k+2 (lanes16-31), VGPR1 K=k+1/k+3
//   C 16x16: VGPR v -> row v (lanes0-15) / row v+8 (lanes16-31), col = lane&15
// A (=F, 8MB) and B (=W1, 0.5MB) re-reads across tiles are served by the
// 192MB L2, so HBM traffic stays ~one copy of each operand.
// ---------------------------------------------------------------------------
__global__ void gemm_wmma_f32(const float* __restrict__ A,
                              const float* __restrict__ B,
                              const float* __restrict__ bias,
                              float* __restrict__ out,
                              int M, int Nn, int Kk, int relu)
{
    int wave = (blockIdx.x * blockDim.x + threadIdx.x) >> 5;
    int lane = threadIdx.x & 31;
    int tilesN = Nn >> 4;
    int tilesTot = (M >> 4) * tilesN;
    if (wave >= tilesTot) return;          // wave-uniform: EXEC stays all-ones
    int tm = wave / tilesN, tn = wave % tilesN;
    int r0 = tm << 4, c0 = tn << 4;
    int hl = lane & 15;                    // row within A frag / col within B frag
    int koff = (lane >> 4) << 1;           // 0 for lanes 0-15, 2 for lanes 16-31

    v8f acc = {};
    for (int k = 0; k < Kk; k += 4) {
        const float* ap = A + (size_t)(r0 + hl) * Kk + (k + koff);
        v2f a; a.x = ap[0]; a.y = ap[1];
        const float* bp = B + (size_t)(k + koff) * Nn + (c0 + hl);
        v2f b; b.x = bp[0]; b.y = bp[Nn];
        acc = __builtin_amdgcn_wmma_f32_16x16x4_f32(
            false, a, false, b, (short)0, acc, false, false);
    }
    int rbase = r0 + ((lane >> 4) << 3);
    float bs = bias[c0 + hl];
#pragma unroll
    for (int v = 0; v < 8; ++v) {
        float x = acc[v] + bs;
        if (relu) x = fmaxf(x, 0.0f);
        out[(size_t)(rbase + v) * Nn + (c0 + hl)] = x;
    }
}

// ---------------------------------------------------------------------------
// Graph structure. Single pass over the 64MB adjacency:
//   degree_kernel reads A once, emitting per-row degree, D^-1/2, and a packed
//   2MB edge bitmask (ballot per 32-col chunk). fill_kernel then builds the
//   CSR from the bitmask only (order-preserving, deterministic, no atomics).
// ---------------------------------------------------------------------------
__global__ void degree_kernel(const float* __restrict__ A,
                              unsigned* __restrict__ bits,
                              unsigned* __restrict__ cnt,
                              float* __restrict__ dis, int n)
{
    int wid  = (blockIdx.x * blockDim.x + threadIdx.x) >> 5;
    int lane = threadIdx.x & 31;
    if (wid >= n) return;
    const float* row = A + (size_t)wid * n;
    unsigned* brow = bits + (size_t)wid * (n >> 5);
    int c = 0;
    for (int j0 = 0; j0 < n; j0 += 32) {
        int j = j0 + lane;
        bool p = (row[j] != 0.0f) && (j != wid);
        unsigned m32 = (unsigned)__ballot(p);
        if (lane == 0) brow[j0 >> 5] = m32;
        c += __popc(m32);                  // wave-uniform running count
    }
    if (lane == 0) {
        cnt[wid] = (unsigned)c;
        dis[wid] = rsqrtf((float)c + 1.0f);   // D^-1/2 with self loop
    }
}

__global__ void scan_kernel(const unsigned* __restrict__ cnt,
                            unsigned* __restrict__ rp, int n)
{
    if (blockIdx.x == 0 && threadIdx.x == 0) {
        unsigned s = 0;
        for (int i = 0; i < n; ++i) { rp[i] = s; s += cnt[i]; }
        rp[n] = s;
    }
}

__global__ void fill_kernel(const unsigned* __restrict__ bits,
                            const unsigned* __restrict__ rp,
                            const float* __restrict__ dis,
                            unsigned* __restrict__ col,
                            unsigned* __restrict__ erow,
                            float* __restrict__ atv,
                            int n, unsigned cap)
{
    int wid  = (blockIdx.x * blockDim.x + threadIdx.x) >> 5;
    int lane = threadIdx.x & 31;
    if (wid >= n) return;
    const unsigned* brow = bits + (size_t)wid * (n >> 5);
    unsigned base = rp[wid];
    float di = dis[wid];
    for (int j0 = 0; j0 < n; j0 += 32) {
        unsigned m32 = brow[j0 >> 5];      // broadcast load (wave-uniform)
        bool p = (m32 >> lane) & 1u;
        unsigned off = __popc(m32 & ((1u << lane) - 1u));
        if (p) {
            unsigned e = base + off;
            int j = j0 + lane;
            if (e < cap) {
                col[e]  = (unsigned)j;
                erow[e] = (unsigned)wid;
                atv[e]  = di * dis[j];      // D^-1/2 A D^-1/2 edge value
            }
        }
        base += __popc(m32);
    }
}

// ---------------------------------------------------------------------------
// Per-iteration kernels
// ---------------------------------------------------------------------------
__global__ void funit_kernel(const float* __restrict__ Fk,
                             const float* __restrict__ dis,
                             float* __restrict__ Funit, int n)
{
    int i = blockIdx.x * blockDim.x + threadIdx.x;
    if (i >= n) return;
    float s = dis[i];
    float v[CC]; float nrm = 0.0f;
#pragma unroll
    for (int c = 0; c < CC; ++c) { v[c] = Fk[i * CC + c] * s; nrm += v[c] * v[c]; }
    nrm = fmaxf(sqrtf(nrm), 1e-8f);
    float inv = 1.0f / nrm;
#pragma unroll
    for (int c = 0; c < CC; ++c) Funit[i * CC + c] = v[c] * inv;
}

__global__ void edgey_kernel(const float* __restrict__ Funit,
                             const unsigned* __restrict__ col,
                             const unsigned* __restrict__ erow,
                             const unsigned* __restrict__ rp,
                             float* __restrict__ ey, int n, unsigned cap)
{
    unsigned m = rp[n]; if (m > cap) m = cap;
    unsigned stride = gridDim.x * blockDim.x;
    for (unsigned e = blockIdx.x * blockDim.x + threadIdx.x; e < m; e += stride) {
        unsigned i = erow[e], j = col[e];
        const float4* u = (const float4*)(Funit + (size_t)i * CC);
        const float4* w = (const float4*)(Funit + (size_t)j * CC);
        float d = 0.0f;
#pragma unroll
        for (int t = 0; t < 4; ++t) {
            float4 a = u[t], b = w[t];
            d += a.x * b.x + a.y * b.y + a.z * b.z + a.w * b.w;
        }
        float y = 1.0f - d;
        ey[e] = fminf(fmaxf(y, 0.0f), 2.0f);
    }
}

// qs layout: [0,1]=prefix(lo,hi) [2,3]=rank-remaining [4,5]=rank lo/hi [6]=m
// qf layout: [0]=frac [1]=lam_k
__global__ void qinit_kernel(const unsigned* __restrict__ rp, int n,
                             unsigned cap, unsigned* __restrict__ qs,
                             float* __restrict__ qf)
{
    if (blockIdx.x == 0 && threadIdx.x == 0) {
        unsigned m = rp[n]; if (m > cap) m = cap;
        float pos = QQ * ((float)m - 1.0f);       // f32, matches reference
        unsigned lo = (unsigned)floorf(pos);
        unsigned hi = (unsigned)ceilf(pos);
        qs[4] = lo; qs[5] = hi; qs[6] = m;
        qf[0] = pos - (float)lo;
    }
}

__global__ void qreset_kernel(unsigned* __restrict__ qs)
{
    if (blockIdx.x == 0 && threadIdx.x == 0) {
        qs[0] = 0u; qs[1] = 0u; qs[2] = qs[4]; qs[3] = qs[5];
    }
}

__global__ void zerohist_kernel(unsigned* __restrict__ hist)
{
    if (threadIdx.x < 512) hist[threadIdx.x] = 0u;
}

// MSD radix histogram over f32 bit patterns (all y >= 0 -> order-preserving).
// LDS-privatized per block (64-bank ds atomics), merged once into global.
__global__ void rhist_kernel(const float* __restrict__ ey,
                             const unsigned* __restrict__ qs,
                             unsigned* __restrict__ hist, int shift)
{
    __shared__ unsigned lh[512];
    for (int t = threadIdx.x; t < 512; t += blockDim.x) lh[t] = 0u;
    __syncthreads();

    unsigned m = qs[6];
    unsigned hm = (shift == 24) ? 0u : (0xFFFFFFFFu << (shift + 8));
    unsigned p0 = qs[0] & hm, p1 = qs[1] & hm;
    unsigned stride = gridDim.x * blockDim.x;
    for (unsigned e = blockIdx.x * blockDim.x + threadIdx.x; e < m; e += stride) {
        unsigned b = __float_as_uint(ey[e]);
        unsigned byte = (b >> shift) & 255u;
        unsigned hb = b & hm;
        if (hb == p0) atomicAdd(&lh[byte], 1u);
        if (hb == p1) atomicAdd(&lh[256 + byte], 1u);
    }
    __syncthreads();
    for (int t = threadIdx.x; t < 512; t += blockDim.x) {
        unsigned v = lh[t];
        if (v) atomicAdd(&hist[t], v);
    }
}

__global__ void rpick_kernel(unsigned* __restrict__ qs,
                             const unsigned* __restrict__ hist, int shift)
{
    if (blockIdx.x != 0 || threadIdx.x != 0) return;
    for (int s = 0; s < 2; ++s) {
        unsigned r = qs[2 + s], cum = 0;
        const unsigned* h = hist + 256 * s;
        for (unsigned b = 0; b < 256; ++b) {
            unsigned c = h[b];
            if (r < cum + c) {
                qs[s] |= (b << shift);
                qs[2 + s] = r - cum;
                break;
            }
            cum += c;
        }
    }
}

__global__ void qfinal_kernel(const unsigned* __restrict__ qs,
                              float* __restrict__ qf)
{
    if (blockIdx.x == 0 && threadIdx.x == 0) {
        float vlo = __uint_as_float(qs[0]);
        float vhi = __uint_as_float(qs[1]);
        float frac = qf[0];
        float gamma = fmaxf(vlo * (1.0f - frac) + vhi * frac, EPS_F);
        qf[1] = gamma / SCAD_A;               // lam_k
    }
}

// SCAD IRLS weight per edge, times A_tilde
__global__ void edgew_kernel(const float* __restrict__ ey,
                             const float* __restrict__ atv,
                             const unsigned* __restrict__ qs,
                             const float* __restrict__ qf,
                             float* __restrict__ wt)
{
    unsigned m = qs[6];
    float lam = qf[1];
    float alam = SCAD_A * lam;
    unsigned stride = gridDim.x * blockDim.x;
    for (unsigned e = blockIdx.x * blockDim.x + threadIdx.x; e < m; e += stride) {
        float y = ey[e];
        float ys = fmaxf(y, EPS_F);
        float wmid = fmaxf((alam - ys) / ((SCAD_A - 1.0f) * ys), 0.0f);
        float w = (y <= lam) ? 1.0f : ((y <= alam) ? wmid : 0.0f);
        wt[e] = w * atv[e];
    }
}

__global__ void denom_kernel(const unsigned* __restrict__ rp,
                             const float* __restrict__ wt,
                             float* __restrict__ denom, int n, unsigned cap)
{
    int i = blockIdx.x * blockDim.x + threadIdx.x;
    if (i >= n) return;
    unsigned e0 = rp[i], e1 = rp[i + 1];
    if (e0 > cap) e0 = cap;
    if (e1 > cap) e1 = cap;
    float s = LAM_F;
    for (unsigned e = e0; e < e1; ++e) s += wt[e];
    denom[i] = s;
}

// QN-IRLS step:  Fk_next = (Wt @ Fk + lam * F0) / denom
// Fk is 256KB -> random col gathers hit L2.
__global__ void spmm_kernel(const unsigned* __restrict__ rp,
                            const unsigned* __restrict__ col,
                            const float* __restrict__ wt,
                            const float* __restrict__ denom,
                            const float* __restrict__ Fin,
                            const float* __restrict__ F0,
                            float* __restrict__ Fout, int n, unsigned cap)
{
    int t = blockIdx.x * blockDim.x + threadIdx.x;
    int row = t >> 4, c = t & 15;
    if (row >= n) return;
    unsigned e0 = rp[row], e1 = rp[row + 1];
    if (e0 > cap) e0 = cap;
    if (e1 > cap) e1 = cap;
    float acc = LAM_F * F0[(size_t)row * CC + c];
    for (unsigned e = e0; e < e1; ++e)
        acc += wt[e] * Fin[(size_t)col[e] * CC + c];
    Fout[(size_t)row * CC + c] = acc / denom[row];
}

// ---------------------------------------------------------------------------
// Host launcher
// ---------------------------------------------------------------------------
extern "C" void kernel_launch(void* const* d_in, const int* in_sizes, int n_in,
                              void* d_out, int out_size, void* d_ws, size_t ws_size,
                              hipStream_t stream)
{
    const float* A  = (const float*)d_in[0];
    const float* F  = (const float*)d_in[1];
    const float* W1 = (const float*)d_in[2];
    const float* b1 = (const float*)d_in[3];
    const float* W2 = (const float*)d_in[4];
    const float* b2 = (const float*)d_in[5];

    // ---- carve workspace ----
    char* wsb = (char*)d_ws;
    size_t off = 0;
    auto carve = [&](size_t bytes) -> char* {
        off = (off + 255) & ~(size_t)255;
        char* p = wsb + off;
        off += bytes;
        return p;
    };
    unsigned* bits  = (unsigned*)carve((size_t)NN * NW * sizeof(unsigned)); // 2MB
    unsigned* cnt   = (unsigned*)carve(NN * sizeof(unsigned));
    unsigned* rp    = (unsigned*)carve((NN + 1) * sizeof(unsigned));
    float*    dis   = (float*)   carve(NN * sizeof(float));
    float*    H     = (float*)   carve((size_t)NN * HID * sizeof(float));
    float*    F0    = (float*)   carve((size_t)NN * CC * sizeof(float));
    float*    Fb0   = (float*)   carve((size_t)NN * CC * sizeof(float));
    float*    Fb1   = (float*)   carve((size_t)NN * CC * sizeof(float));
    float*    Funit = (float*)   carve((size_t)NN * CC * sizeof(float));
    float*    denom = (float*)   carve(NN * sizeof(float));
    unsigned* hist  = (unsigned*)carve(512 * sizeof(unsigned));
    unsigned* qs    = (unsigned*)carve(8 * sizeof(unsigned));
    float*    qf    = (float*)   carve(4 * sizeof(float));

    // size edge arrays from remaining workspace (expected edges ~67K)
    size_t remain = (ws_size > off + 4096) ? (ws_size - off - 4096) : 0;
    size_t capmax = remain / (5 * sizeof(unsigned));
    unsigned cap = (capmax > (1u << 21)) ? (1u << 21) : (unsigned)capmax;
    if (cap < 1024) cap = 1024;
    unsigned* col  = (unsigned*)carve((size_t)cap * sizeof(unsigned));
    unsigned* erow = (unsigned*)carve((size_t)cap * sizeof(unsigned));
    float*    atv  = (float*)   carve((size_t)cap * sizeof(float));
    float*    ey   = (float*)   carve((size_t)cap * sizeof(float));
    float*    wt   = (float*)   carve((size_t)cap * sizeof(float));

    // ---- MLP via f32 WMMA ----
    // H = relu(F @ W1 + b1): 256x16 tiles = 4096 waves -> 1024 blocks x 4 waves
    gemm_wmma_f32<<<1024, 128, 0, stream>>>(F, W1, b1, H, NN, HID, IN_DIM, 1);
    // F0 = H @ W2 + b2: 256x1 tiles = 256 waves -> 64 blocks x 4 waves
    gemm_wmma_f32<<<64, 128, 0, stream>>>(H, W2, b2, F0, NN, CC, HID, 0);

    // ---- graph structure: single 64MB sweep of A ----
    degree_kernel<<<512, 256, 0, stream>>>(A, bits, cnt, dis, NN);
    scan_kernel<<<1, 1, 0, stream>>>(cnt, rp, NN);
    fill_kernel<<<512, 256, 0, stream>>>(bits, rp, dis, col, erow, atv, NN, cap);
    qinit_kernel<<<1, 1, 0, stream>>>(rp, NN, cap, qs, qf);

    // ---- K QN-IRLS iterations ----
    const float* cur = F0;
    for (int k = 0; k < KSTEPS; ++k) {
        funit_kernel<<<NN / 256, 256, 0, stream>>>(cur, dis, Funit, NN);
        edgey_kernel<<<256, 256, 0, stream>>>(Funit, col, erow, rp, ey, NN, cap);
        // exact dual-rank radix select (ranks floor(pos), ceil(pos))
        qreset_kernel<<<1, 1, 0, stream>>>(qs);
        for (int p = 0; p < 4; ++p) {
            int shift = 24 - 8 * p;
            zerohist_kernel<<<1, 512, 0, stream>>>(hist);
            rhist_kernel<<<128, 256, 0, stream>>>(ey, qs, hist, shift);
            rpick_kernel<<<1, 1, 0, stream>>>(qs, hist, shift);
        }
        qfinal_kernel<<<1, 1, 0, stream>>>(qs, qf);
        edgew_kernel<<<256, 256, 0, stream>>>(ey, atv, qs, qf, wt);
        denom_kernel<<<NN / 256, 256, 0, stream>>>(rp, wt, denom, NN, cap);
        float* nxt = (k == KSTEPS - 1) ? (float*)d_out : ((k & 1) ? Fb1 : Fb0);
        spmm_kernel<<<(NN * CC) / 256, 256, 0, stream>>>(rp, col, wt, denom,
                                                         cur, F0, nxt, NN, cap);
        cur = nxt;
    }
}